// RegressionLoss_19499151524051
// MI455X (gfx1250) — compile-verified
//
#include <hip/hip_runtime.h>
#include <cstdint>
#include <cstddef>

#define BLOCK 256
#define MAXG  256

// ---------------------------------------------------------------------------
// Compile-time constants matching the reference:
//   RATE = 22050/256; LO/HI computed in float64 then cast to float32 (numpy).
// ---------------------------------------------------------------------------
#define RATE_D (22050.0 / 256.0)
__constant__ float c_LO[6] = {
    (float)(0.0        * RATE_D), (float)(0.32537674 * RATE_D),
    (float)(0.47555801 * RATE_D), (float)(0.64588683 * RATE_D),
    (float)(1.16883525 * RATE_D), (float)(2.17128976 * RATE_D)};
__constant__ float c_HI[6] = {
    (float)(0.32537674 * RATE_D), (float)(0.47555801 * RATE_D),
    (float)(0.64588683 * RATE_D), (float)(1.16883525 * RATE_D),
    (float)(2.17128976 * RATE_D), __builtin_huge_valf()};
__constant__ float c_INV[6] = {1.0f, 0.5f, 0.25f, 0.125f, 0.0625f, 0.03125f};

// gfx1250 async global->LDS path (probe via __has_builtin; falls back cleanly)
#if defined(__gfx1250__) && __has_builtin(__builtin_amdgcn_global_load_async_to_lds_b32)
#define HAVE_ASYNC_LDS 1
#else
#define HAVE_ASYNC_LDS 0
#endif

#if HAVE_ASYNC_LDS
// b32 variant expects int-typed pointers: (AS1 int* src, AS3 int* lds, imm, imm)
#define AS1_I32(p) ((__attribute__((address_space(1))) int*)(uintptr_t)(p))
#define AS3_I32(p) ((__attribute__((address_space(3))) int*)(unsigned)(uintptr_t)(p))
#endif

// ---------------------------------------------------------------------------
// Per-block partial kernel. grid = (blocksPerSample, B), block = 256 threads.
// Each block: stage+sort annotations in LDS, then one anchor per thread.
// Writes deterministic per-block partial (sum, count) into workspace.
// ---------------------------------------------------------------------------
__global__ void __launch_bounds__(BLOCK)
fcos_partial_kernel(const float* __restrict__ reg,   // [B][A][2]
                    const float* __restrict__ ann,   // [B][G][3]
                    const float* __restrict__ a0, const float* __restrict__ a1,
                    const float* __restrict__ a2, const float* __restrict__ a3,
                    const float* __restrict__ a4, const float* __restrict__ a5,
                    int n0, int n1, int n2, int n3, int n4, int n5,
                    int A, int G,
                    float*    __restrict__ psum,     // [B][blocksPerSample]
                    unsigned* __restrict__ pcnt)     // [B][blocksPerSample]
{
  __shared__ float2   sBox[MAXG];     // (start, end) -- sorted in place
  __shared__ float    sLen[MAXG];
  __shared__ float    redS[BLOCK / 32];
  __shared__ unsigned redC[BLOCK / 32];

  const int b   = blockIdx.y;
  const int tid = threadIdx.x;
  const int bps = gridDim.x;
  const float* annB = ann + (size_t)b * (size_t)G * 3;

  float s = 0.0f, e = 0.0f;

#if HAVE_ASYNC_LDS
  // ---- stage annotations into LDS with gfx1250 async copies ----
  if (tid < G) {
    __builtin_amdgcn_global_load_async_to_lds_b32(
        AS1_I32(annB + (size_t)tid * 3 + 0), AS3_I32(&sBox[tid].x), 0, 0);
    __builtin_amdgcn_global_load_async_to_lds_b32(
        AS1_I32(annB + (size_t)tid * 3 + 1), AS3_I32(&sBox[tid].y), 0, 0);
  }
#if __has_builtin(__builtin_amdgcn_s_wait_asynccnt)
  __builtin_amdgcn_s_wait_asynccnt(0);
#else
  asm volatile("s_wait_asynccnt 0x0" ::: "memory");
#endif
  __syncthreads();
  if (tid < G) {
    float2 v = sBox[tid];
    s = v.x; e = v.y;
    sLen[tid] = e - s;
  }
#else
  // ---- plain staging fallback ----
  if (tid < G) {
    s = annB[(size_t)tid * 3 + 0];
    e = annB[(size_t)tid * 3 + 1];
    sLen[tid] = e - s;
  }
#endif
  __syncthreads();   // all unsorted reads done; lengths published

  // ---- stable rank sort by length (matches jnp.argsort stable) ----
  if (tid < G) {
    const float len = e - s;
    int r = 0;
    for (int k = 0; k < G; ++k) {
      const float lk = sLen[k];
      r += (lk < len) || ((lk == len) && (k < tid)) ? 1 : 0;
    }
    sBox[r] = make_float2(s, e);   // ranks are a permutation (stable ties)
  }
  __syncthreads();

  // ---- one anchor per thread ----
  const int a = blockIdx.x * BLOCK + tid;
  float    loss = 0.0f;
  unsigned cnt  = 0;

  if (a < A) {
    // locate level + local index
    int lvl, rem = a;
    const float* ap;
    if (rem < n0)              { lvl = 0; ap = a0; }
    else { rem -= n0;
      if (rem < n1)            { lvl = 1; ap = a1; }
      else { rem -= n1;
        if (rem < n2)          { lvl = 2; ap = a2; }
        else { rem -= n2;
          if (rem < n3)        { lvl = 3; ap = a3; }
          else { rem -= n3;
            if (rem < n4)      { lvl = 4; ap = a4; }
            else { rem -= n4;    lvl = 5; ap = a5; } } } } }

    const float pt = ap[rem];
    const float lo = c_LO[lvl];
    const float hi = c_HI[lvl];

    const float* rp = reg + ((size_t)b * (size_t)A + (size_t)a) * 2;
    __builtin_prefetch(rp, 0, 1);      // global_prefetch_b8 on gfx1250

    // scan sorted boxes; first match == argmax(mask) in sorted order
    bool  found = false;
    float as = 0.0f, ae = 0.0f;
    for (int k = 0; k < G; ++k) {
      const float2 box = sBox[k];      // broadcast ds_load_b64
      const float l = pt - box.x;
      const float r = box.y - pt;
      const float m = fmaxf(l, r);
      const bool match = (l >= 0.0f) && (r >= 0.0f) && (m >= lo) && (m < hi);
      if (match && !found) { found = true; as = box.x; ae = box.y; }
      if ((k & 15) == 15) {
        // wave32 early exit once every lane has its first match
        if (__builtin_amdgcn_ballot_w32(!found) == 0u) break;
      }
    }

    if (found) {
      const float inv = c_INV[lvl];
      const float A0 = as * inv, A1 = ae * inv;   // exact: divide by 2^lvl
      const float B0 = rp[0],    B1 = rp[1];
      float inter = fminf(A1, B1) - fmaxf(A0, B0);
      inter = fmaxf(inter, 0.0f);
      const float uni = (A1 - A0) + (B1 - B0) - inter;
      const float enc = fmaxf(A1, B1) - fminf(A0, B0);
      const float iou  = inter / (uni + 1e-7f);
      const float giou = iou - (enc - uni) / (enc + 1e-7f);
      loss = 1.0f - giou;
      cnt  = 1u;
    }
  }

  // ---- deterministic block reduction (wave32 shfl tree + LDS) ----
  for (int off = 16; off > 0; off >>= 1) {
    loss += __shfl_down(loss, off, 32);
    cnt  += __shfl_down(cnt,  off, 32);
  }
  const int wave = tid >> 5;
  const int lane = tid & 31;
  if (lane == 0) { redS[wave] = loss; redC[wave] = cnt; }
  __syncthreads();
  if (tid == 0) {
    float    ts = 0.0f;
    unsigned tc = 0;
    for (int w = 0; w < BLOCK / 32; ++w) { ts += redS[w]; tc += redC[w]; }
    psum[(size_t)b * bps + blockIdx.x] = ts;
    pcnt[(size_t)b * bps + blockIdx.x] = tc;
  }
}

// ---------------------------------------------------------------------------
// Final reduction: fixed-order sum of per-block partials -> deterministic.
// ---------------------------------------------------------------------------
__global__ void fcos_final_kernel(const float* __restrict__ psum,
                                  const unsigned* __restrict__ pcnt,
                                  float* __restrict__ out, int B, int bps)
{
  const int b = blockIdx.x * blockDim.x + threadIdx.x;
  if (b < B) {
    const float*    ps = psum + (size_t)b * bps;
    const unsigned* pc = pcnt + (size_t)b * bps;
    float    s = 0.0f;
    unsigned c = 0;
    for (int i = 0; i < bps; ++i) { s += ps[i]; c += pc[i]; }
    out[b] = s / (float)(c > 0u ? c : 1u);
  }
}

// ---------------------------------------------------------------------------
// Host launcher
// ---------------------------------------------------------------------------
extern "C" void kernel_launch(void* const* d_in, const int* in_sizes, int n_in,
                              void* d_out, int out_size, void* d_ws, size_t ws_size,
                              hipStream_t stream) {
  const float* reg = (const float*)d_in[0];
  const float* ann = (const float*)d_in[1];
  const float* a0  = (const float*)d_in[2];
  const float* a1  = (const float*)d_in[3];
  const float* a2  = (const float*)d_in[4];
  const float* a3  = (const float*)d_in[5];
  const float* a4  = (const float*)d_in[6];
  const float* a5  = (const float*)d_in[7];

  const int n0 = in_sizes[2], n1 = in_sizes[3], n2 = in_sizes[4];
  const int n3 = in_sizes[5], n4 = in_sizes[6], n5 = in_sizes[7];
  const int A  = n0 + n1 + n2 + n3 + n4 + n5;          // 32256
  const int B  = in_sizes[0] / (A * 2);                // 8
  int G = in_sizes[1] / (B * 3);                       // 256
  if (G > MAXG) G = MAXG;

  const int bps = (A + BLOCK - 1) / BLOCK;             // 126

  float*    psum = (float*)d_ws;
  unsigned* pcnt = (unsigned*)((char*)d_ws + sizeof(float) * (size_t)B * bps);

  dim3 grid(bps, B);
  fcos_partial_kernel<<<grid, BLOCK, 0, stream>>>(
      reg, ann, a0, a1, a2, a3, a4, a5,
      n0, n1, n2, n3, n4, n5, A, G, psum, pcnt);

  fcos_final_kernel<<<1, 64, 0, stream>>>(psum, pcnt, (float*)d_out, B, bps);
}